// Yolov3Loss_9070970929646
// MI455X (gfx1250) — compile-verified
//
#include <hip/hip_runtime.h>
#include <hip/hip_bf16.h>
#include <math.h>

#define G_BOXES 120
#define NB_MAX  256
#define TPB     256
#define WPB     (TPB / 32)

typedef __attribute__((ext_vector_type(2))) float v2f;
typedef __attribute__((ext_vector_type(8))) float v8f;

// ANCHORS flattened (9 x 2)
__constant__ float c_anch[18] = {10,13, 16,30, 33,23, 30,61, 62,45, 59,119,
                                 116,90, 156,198, 373,326};

// Monotonic (order-preserving) uint key for float: a > b  <=>  fkey(a) > fkey(b)
__device__ inline unsigned fkey(float f) {
  unsigned u = __float_as_uint(f);
  return (u & 0x80000000u) ? ~u : (u | 0x80000000u);
}

// noinline so pass A and pass B execute the exact same instruction sequence
// (bit-identical IoU values -> float-equality "assign" test matches pass A's max).
__device__ __attribute__((noinline))
float4 decode_box(const float* raw, int xi, int yi, int g, int scale, int an) {
  float invg = 1.0f / (float)g;
  float sx = 1.0f / (1.0f + __expf(-raw[0]));
  float sy = 1.0f / (1.0f + __expf(-raw[1]));
  float4 r;
  r.x = (sx + (float)xi) * invg;
  r.y = (sy + (float)yi) * invg;
  r.z = __expf(raw[2]) * c_anch[(scale * 3 + an) * 2 + 0] * (1.0f / 608.0f);
  r.w = __expf(raw[3]) * c_anch[(scale * 3 + an) * 2 + 1] * (1.0f / 608.0f);
  return r;
}

__device__ __attribute__((noinline))
float iou_fn(float px, float py, float pw, float ph, const float* gb) {
  float gx = gb[0], gy = gb[1], gw = gb[2], gh = gb[3];
  float px1 = px - pw * 0.5f, px2 = px + pw * 0.5f;
  float py1 = py - ph * 0.5f, py2 = py + ph * 0.5f;
  float gx1 = gx - gw * 0.5f, gx2 = gx + gw * 0.5f;
  float gy1 = gy - gh * 0.5f, gy2 = gy + gh * 0.5f;
  float iw = fminf(gx2, px2) - fmaxf(gx1, px1);
  float ih = fminf(gy2, py2) - fmaxf(gy1, py1);
  float inter = iw * ih;                 // no clamp (faithful to reference)
  float uni = gw * gh + pw * ph;
  return inter / (uni - inter);
}

// Cross-lane accumulation on the matrix pipe:
// D = Ones(16x4) * B(4x16) + C  =>  every row of D holds column sums of B.
// Total over all lanes' (b.x + b.y) is invariant to the exact B striping.
__device__ inline v8f wsum(float val, v8f c) {
  v2f a; a.x = 1.0f; a.y = 1.0f;
  v2f b; b.x = val;  b.y = 0.0f;
  return __builtin_amdgcn_wmma_f32_16x16x4_f32(false, a, false, b, (short)0, c,
                                               false, false);
}

// After accumulation, lane L holds colsum[L % 16] in acc[0]; sum the 16
// distinct column sums with a 4-step xor-shuffle (all lanes end with total).
__device__ inline float wave_total(v8f acc) {
  float v = acc[0];
  v += __shfl_xor(v, 1);
  v += __shfl_xor(v, 2);
  v += __shfl_xor(v, 4);
  v += __shfl_xor(v, 8);
  return v;
}

__global__ void zero_keys(unsigned* mk) {
  if (threadIdx.x < 3 * 128) mk[threadIdx.x] = 0u;
}

// Pass A: full-bandwidth pass. Per cell: obj-BCE, cls-BCE (80 ch), sum(obj),
// and per-GT-box IoU max (wave shfl-max -> LDS atomicMax -> global atomicMax).
__global__ void __launch_bounds__(TPB)
passA(const float* __restrict__ yo, const float* __restrict__ yt,
      const float* __restrict__ gtb, unsigned* __restrict__ maxkey,
      float* __restrict__ partials, int g, int scale, int N) {
  __shared__ float    sgt[G_BOXES * 4];
  __shared__ unsigned smax[G_BOXES];
  __shared__ float    sred[WPB * 3];

  for (int t = threadIdx.x; t < G_BOXES * 4; t += blockDim.x) sgt[t] = gtb[t];
  for (int t = threadIdx.x; t < G_BOXES; t += blockDim.x) smax[t] = 0u;
  __syncthreads();

  const int lane = threadIdx.x & 31;
  const int wib  = threadIdx.x >> 5;
  const int gwav = blockIdx.x * WPB + wib;
  const int twav = gridDim.x * WPB;

  v8f aObj = {}; v8f aCls = {}; v8f aO = {};

  for (int i0 = gwav * 32; i0 < N; i0 += twav * 32) {   // wave-uniform bound
    int idx = i0 + lane;
    float validf = (idx < N) ? 1.0f : 0.0f;
    int ci = (idx < N) ? idx : (N - 1);
    int an = ci % 3;  int cell = ci / 3;
    int xi = cell % g; int t2 = cell / g; int yi = t2 % g;
    const float* raw = yo + (size_t)cell * 255 + (size_t)an * 85;
    const float* tr  = yt + (size_t)cell * 255 + (size_t)an * 85;

    float4 box = decode_box(raw, xi, yi, g, scale, an);

    // Reference applies BCE-with-logits to the *sigmoided* values.
    float xo   = 1.0f / (1.0f + __expf(-raw[4]));
    float tobj = tr[4];
    float bobj = xo * (1.0f - tobj) + __logf(1.0f + __expf(-xo));  // xo >= 0
    float bcls = 0.0f;
    for (int c = 0; c < 80; ++c) {
      float xc = 1.0f / (1.0f + __expf(-raw[5 + c]));
      float tc = tr[5 + c];
      bcls += xc * (1.0f - tc) + __logf(1.0f + __expf(-xc));
    }

    // matrix-pipe accumulation (EXEC is all-ones here: uniform control flow)
    aObj = wsum(bobj * validf, aObj);
    aCls = wsum(bcls * validf, aCls);
    aO   = wsum(tobj * validf, aO);

    for (int gi = 0; gi < G_BOXES; ++gi) {
      float iou = iou_fn(box.x, box.y, box.z, box.w, &sgt[gi * 4]);
      unsigned k = (idx < N) ? fkey(iou) : 0u;
      unsigned o;
      o = __shfl_xor(k, 16); if (o > k) k = o;
      o = __shfl_xor(k, 8);  if (o > k) k = o;
      o = __shfl_xor(k, 4);  if (o > k) k = o;
      o = __shfl_xor(k, 2);  if (o > k) k = o;
      o = __shfl_xor(k, 1);  if (o > k) k = o;
      if (lane == 0) atomicMax(&smax[gi], k);
    }
  }

  float tO0 = wave_total(aObj);
  float tC0 = wave_total(aCls);
  float tS0 = wave_total(aO);
  if (lane == 0) {
    sred[wib * 3 + 0] = tO0; sred[wib * 3 + 1] = tC0; sred[wib * 3 + 2] = tS0;
  }
  __syncthreads();
  if (threadIdx.x < G_BOXES) atomicMax(&maxkey[threadIdx.x], smax[threadIdx.x]);
  if (threadIdx.x == 0) {
    float s0 = 0.f, s1 = 0.f, s2 = 0.f;
    for (int w = 0; w < WPB; ++w) {
      s0 += sred[w * 3 + 0]; s1 += sred[w * 3 + 1]; s2 += sred[w * 3 + 2];
    }
    partials[(size_t)blockIdx.x * 3 + 0] = s0;
    partials[(size_t)blockIdx.x * 3 + 1] = s1;
    partials[(size_t)blockIdx.x * 3 + 2] = s2;
  }
}

// Pass B: light pass (9 of 85 channels). assign via key equality vs global max.
__global__ void __launch_bounds__(TPB)
passB(const float* __restrict__ yo, const float* __restrict__ yt,
      const float* __restrict__ gtb, const unsigned* __restrict__ maxkey,
      float* __restrict__ partials, int g, int scale, int N) {
  __shared__ float    sgt[G_BOXES * 4];
  __shared__ unsigned smk[G_BOXES];
  __shared__ float    sred[WPB * 3];

  for (int t = threadIdx.x; t < G_BOXES * 4; t += blockDim.x) sgt[t] = gtb[t];
  for (int t = threadIdx.x; t < G_BOXES; t += blockDim.x) smk[t] = maxkey[t];
  __syncthreads();

  const int lane = threadIdx.x & 31;
  const int wib  = threadIdx.x >> 5;
  const int gwav = blockIdx.x * WPB + wib;
  const int twav = gridDim.x * WPB;

  v8f aXY = {}; v8f aWH = {}; v8f aNO = {};

  for (int i0 = gwav * 32; i0 < N; i0 += twav * 32) {
    int idx = i0 + lane;
    float validf = (idx < N) ? 1.0f : 0.0f;
    int ci = (idx < N) ? idx : (N - 1);
    int an = ci % 3;  int cell = ci / 3;
    int xi = cell % g; int t2 = cell / g; int yi = t2 % g;
    const float* raw = yo + (size_t)cell * 255 + (size_t)an * 85;
    const float* tr  = yt + (size_t)cell * 255 + (size_t)an * 85;

    float4 box = decode_box(raw, xi, yi, g, scale, an);
    float tx = tr[0], ty = tr[1], tw = tr[2], th = tr[3], tobj = tr[4];

    unsigned anyhit = 0u;
    for (int gi = 0; gi < G_BOXES; ++gi) {
      float iou = iou_fn(box.x, box.y, box.z, box.w, &sgt[gi * 4]);
      anyhit |= (fkey(iou) == smk[gi]) ? 1u : 0u;
    }
    float af  = (anyhit && (idx < N)) ? 1.0f : 0.0f;
    float bls = 2.0f - tw * th;
    float wgt = af * bls;
    float dx = tx - box.x, dy = ty - box.y;
    float dw = tw - box.z, dh = th - box.w;
    float vxy = (dx * dx + dy * dy) * wgt;
    float vwh = (dw * dw + dh * dh) * wgt;
    float vno = (1.0f - af) * (1.0f - tobj) * validf;

    aXY = wsum(vxy, aXY);
    aWH = wsum(vwh, aWH);
    aNO = wsum(vno, aNO);
  }

  float t0 = wave_total(aXY);
  float t1 = wave_total(aWH);
  float t2v = wave_total(aNO);
  if (lane == 0) {
    sred[wib * 3 + 0] = t0; sred[wib * 3 + 1] = t1; sred[wib * 3 + 2] = t2v;
  }
  __syncthreads();
  if (threadIdx.x == 0) {
    float s0 = 0.f, s1 = 0.f, s2 = 0.f;
    for (int w = 0; w < WPB; ++w) {
      s0 += sred[w * 3 + 0]; s1 += sred[w * 3 + 1]; s2 += sred[w * 3 + 2];
    }
    partials[(size_t)blockIdx.x * 3 + 0] = s0;
    partials[(size_t)blockIdx.x * 3 + 1] = s1;
    partials[(size_t)blockIdx.x * 3 + 2] = s2;
  }
}

// Deterministic fixed-order reduction of per-block partials -> final loss.
__global__ void __launch_bounds__(256)
finalize(const float* __restrict__ pA, const float* __restrict__ pB,
         float* __restrict__ out, int nb0, int nb1, int nb2) {
  __shared__ float red[256];
  __shared__ float totals[18];
  int nbs0 = nb0, nbs1 = nb1, nbs2 = nb2;
  for (int tgt = 0; tgt < 18; ++tgt) {
    int sc = tgt / 6, j = tgt % 6;
    int n = (sc == 0) ? nbs0 : (sc == 1) ? nbs1 : nbs2;
    const float* src = ((j < 3) ? pA : pB) + (size_t)sc * NB_MAX * 3 + (j % 3);
    float s = 0.0f;
    for (int k = threadIdx.x; k < n; k += 256) s += src[(size_t)k * 3];
    red[threadIdx.x] = s;
    __syncthreads();
    for (int off = 128; off; off >>= 1) {
      if ((int)threadIdx.x < off) red[threadIdx.x] += red[threadIdx.x + off];
      __syncthreads();
    }
    if (threadIdx.x == 0) totals[tgt] = red[0];
    __syncthreads();
  }
  if (threadIdx.x == 0) {
    const int gs0 = 76, gs1 = 38, gs2 = 19;
    const int gss[3] = {gs0, gs1, gs2};
    float loss = 0.0f;
    for (int i = 0; i < 3; ++i) {
      float Nc = 8.0f * (float)gss[i] * (float)gss[i] * 3.0f;
      float ob  = totals[i * 6 + 0] / Nc;              // mean obj BCE
      float cb  = totals[i * 6 + 1] / (Nc * 80.0f);    // mean cls BCE
      float So  = totals[i * 6 + 2];                   // sum(obj_m)
      float Sxy = totals[i * 6 + 3];
      float Swh = totals[i * 6 + 4];
      float Sno = totals[i * 6 + 5];                   // sum((1-assign)(1-obj))
      loss += Sxy * 0.125f + 0.5f * Swh * 0.125f +
              ob * (So + Sno) * 0.125f + So * cb * 0.125f;
    }
    out[0] = loss;
  }
}

extern "C" void kernel_launch(void* const* d_in, const int* in_sizes, int n_in,
                              void* d_out, int out_size, void* d_ws, size_t ws_size,
                              hipStream_t stream) {
  (void)in_sizes; (void)n_in; (void)out_size; (void)ws_size;
  const float* yo[3] = {(const float*)d_in[0], (const float*)d_in[2], (const float*)d_in[4]};
  const float* yt[3] = {(const float*)d_in[1], (const float*)d_in[3], (const float*)d_in[5]};
  const float* gtb   = (const float*)d_in[6];

  // ws layout: [3*128 u32 maxkey][3*NB_MAX*3 f32 partialsA][3*NB_MAX*3 f32 partialsB]
  unsigned* mk = (unsigned*)d_ws;
  float* pA = (float*)d_ws + 3 * 128;
  float* pB = pA + 3 * NB_MAX * 3;

  const int gs[3] = {76, 38, 19};
  zero_keys<<<1, 512, 0, stream>>>(mk);

  int nbs[3];
  for (int i = 0; i < 3; ++i) {
    int N = 8 * gs[i] * gs[i] * 3;
    int nb = (N + TPB - 1) / TPB; if (nb > NB_MAX) nb = NB_MAX;
    nbs[i] = nb;
    passA<<<nb, TPB, 0, stream>>>(yo[i], yt[i], gtb, mk + i * 128,
                                  pA + (size_t)i * NB_MAX * 3, gs[i], i, N);
  }
  for (int i = 0; i < 3; ++i) {
    int N = 8 * gs[i] * gs[i] * 3;
    passB<<<nbs[i], TPB, 0, stream>>>(yo[i], yt[i], gtb, mk + i * 128,
                                      pB + (size_t)i * NB_MAX * 3, gs[i], i, N);
  }
  finalize<<<1, 256, 0, stream>>>(pA, pB, (float*)d_out, nbs[0], nbs[1], nbs[2]);
}